// FinalSqueeze_70609262346474
// MI455X (gfx1250) — compile-verified
//
#include <hip/hip_runtime.h>

typedef __attribute__((ext_vector_type(2))) float v2f;
typedef __attribute__((ext_vector_type(8))) float v8f;

#define GROUP     15
#define T_DIM     42
#define C_DIM     512
#define B_DIM     256
#define CROP_NUM  619
#define CB        32                          // channels per workgroup
#define SLAB_BYTES (C_DIM * T_DIM * 4)        // 86016 bytes per (b,g) slab
#define SEG_BYTES  (CB * T_DIM * 4)           // 5376 bytes staged per g
#define SEG_CHUNKS (SEG_BYTES / 16)           // 336 b128 chunks per g
#define TILE_FLOATS (GROUP * CB * T_DIM)      // 20160 floats = 80640 B LDS
#define N_CHUNK16  (GROUP * SEG_CHUNKS)       // 5040 b128 chunks total
#define NBLOCKS    39                         // 39*16 = 624 >= 619 (loads <= 623 < 630 valid)

__global__ __launch_bounds__(64) void cummean_wmma_kernel(
    const float* __restrict__ x, float* __restrict__ out) {
  __shared__ float tile[TILE_FLOATS];

  const int chunk = blockIdx.x;   // 0..15  channel chunk
  const int b     = blockIdx.y;   // 0..255 batch
  const int tid   = threadIdx.x;  // 0..63  (2 waves)
  const int c0    = chunk * CB;

  // ---- Stage 15 contiguous [32ch x 42t] segments into LDS via async DMA ----
  const unsigned long long xbase =
      (unsigned long long)x + (unsigned long long)b * GROUP * SLAB_BYTES +
      (unsigned long long)c0 * (T_DIM * 4);
  for (int idx = tid; idx < N_CHUNK16; idx += 64) {
    unsigned g   = (unsigned)idx / SEG_CHUNKS;
    unsigned off = ((unsigned)idx % SEG_CHUNKS) * 16u;
    unsigned long long gaddr = xbase + (unsigned long long)g * SLAB_BYTES + off;
    // Derive the DS address from tile's own address: low 32 bits of the
    // generic pointer are the wave-relative LDS byte offset, and the
    // ptrtoint captures `tile` so the "memory" clobber makes this asm a
    // potential writer of `tile` (keeps the later ds_loads alive).
    const float* lp = &tile[g * (CB * T_DIM) + (off >> 2)];
    unsigned laddr = (unsigned)(unsigned long long)lp;
    asm volatile("global_load_async_to_lds_b128 %0, %1, off"
                 :: "v"(laddr), "v"(gaddr) : "memory");
  }
  asm volatile("s_wait_asynccnt 0x0" ::: "memory");
  __syncthreads();

  // ---- Per-wave WMMA block scan: 16 channels per wave ----
  const int lane = tid & 31;
  const int w    = tid >> 5;     // wave id -> channel sub-group
  const int h    = lane >> 4;    // lane half
  const int n    = lane & 15;    // column N (channel) / row M for A
  const int cc   = w * 16 + n;   // channel within the 32-ch chunk
  const int ccT  = cc * T_DIM;

  // A = 16x16 lower-triangular ones, consumed as four 16x4 slices.
  // A layout (f32 16x4): VGPR0 holds K = k+2h, VGPR1 holds K = k+1+2h; M = n.
  v2f A[4];
#pragma unroll
  for (int kq = 0; kq < 4; ++kq) {
    int k = kq * 4;
    A[kq].x = ((k + 2 * h)     <= n) ? 1.0f : 0.0f;
    A[kq].y = ((k + 2 * h + 1) <= n) ? 1.0f : 0.0f;
  }

  float carry = 0.0f;
  // Store pointer for rows i = base + r + 8h; advanced by 16 rows per block.
  float* p = out + (unsigned long long)b * CROP_NUM * C_DIM + (c0 + cc) +
             (unsigned long long)(8 * h) * C_DIM;
  // Running counts (i+1) as floats; exact integer adds in fp32.
  float cnt[8];
#pragma unroll
  for (int r = 0; r < 8; ++r) cnt[r] = (float)(r + 8 * h + 1);

  for (int blk = 0; blk < NBLOCKS - 1; ++blk) {
    const int base = blk * 16;

    v8f acc;
#pragma unroll
    for (int r = 0; r < 8; ++r) acc[r] = carry;  // C = carry broadcast to all rows

#pragma unroll
    for (int kq = 0; kq < 4; ++kq) {
      // B layout (f32 4x16): lane halves split K; VGPR0 = K0|K2, VGPR1 = K1|K3.
      unsigned i0 = (unsigned)(base + kq * 4 + 2 * h);
      unsigned t0 = i0 / 15u, g0 = i0 - t0 * 15u;
      unsigned i1 = i0 + 1u;
      unsigned t1 = i1 / 15u, g1 = i1 - t1 * 15u;
      v2f Bm;
      Bm.x = tile[g0 * (CB * T_DIM) + ccT + t0];
      Bm.y = tile[g1 * (CB * T_DIM) + ccT + t1];
      acc = __builtin_amdgcn_wmma_f32_16x16x4_f32(
          /*neg_a=*/false, A[kq], /*neg_b=*/false, Bm,
          /*c_mod=*/(short)0, acc, /*reuse_a=*/false, /*reuse_b=*/false);
    }

    // Unpredicated stores (all rows < 619 for blocks 0..37), immediate offsets.
#pragma unroll
    for (int r = 0; r < 8; ++r) {
      p[r * C_DIM] = acc[r] * __builtin_amdgcn_rcpf(cnt[r]);
      cnt[r] += 16.0f;
    }
    // New carry = total through row 15 = acc[7] on lanes 16..31.
    carry = __shfl(acc[7], 16 + n, 32);
    p += 16 * C_DIM;
  }

  // ---- Last (partial) block: base = 608, rows 608..623, store only i < 619 ----
  {
    const int base = (NBLOCKS - 1) * 16;
    v8f acc;
#pragma unroll
    for (int r = 0; r < 8; ++r) acc[r] = carry;
#pragma unroll
    for (int kq = 0; kq < 4; ++kq) {
      unsigned i0 = (unsigned)(base + kq * 4 + 2 * h);
      unsigned t0 = i0 / 15u, g0 = i0 - t0 * 15u;
      unsigned i1 = i0 + 1u;
      unsigned t1 = i1 / 15u, g1 = i1 - t1 * 15u;
      v2f Bm;
      Bm.x = tile[g0 * (CB * T_DIM) + ccT + t0];
      Bm.y = tile[g1 * (CB * T_DIM) + ccT + t1];
      acc = __builtin_amdgcn_wmma_f32_16x16x4_f32(
          false, A[kq], false, Bm, (short)0, acc, false, false);
    }
#pragma unroll
    for (int r = 0; r < 8; ++r) {
      int i = base + r + 8 * h;
      if (i < CROP_NUM) {
        p[r * C_DIM] = acc[r] * __builtin_amdgcn_rcpf(cnt[r]);
      }
    }
  }
}

extern "C" void kernel_launch(void* const* d_in, const int* in_sizes, int n_in,
                              void* d_out, int out_size, void* d_ws, size_t ws_size,
                              hipStream_t stream) {
  (void)in_sizes; (void)n_in; (void)out_size; (void)d_ws; (void)ws_size;
  const float* x = (const float*)d_in[0];
  float* out = (float*)d_out;
  dim3 grid(C_DIM / CB, B_DIM);   // (16, 256) = 4096 workgroups
  cummean_wmma_kernel<<<grid, 64, 0, stream>>>(x, out);
}